// S4_55422257987929
// MI455X (gfx1250) — compile-verified
//
#include <hip/hip_runtime.h>
#include <hip/hip_bf16.h>
#include <cstdint>

// ---------------------------------------------------------------------------
// S4D layer for MI455X (gfx1250): chunked state-space scan, all matmuls via
// V_WMMA_F32_16X16X4_F32 (wave32 WMMA, fp32 end-to-end), LDS staging via
// GLOBAL_LOAD_ASYNC_TO_LDS_B128 when the toolchain exposes it.
// B=16, H=512, N=64, L=2048, C=1, chunk T=64, J=32 chunks.
// ---------------------------------------------------------------------------

typedef __attribute__((ext_vector_type(2))) float v2f;
typedef __attribute__((ext_vector_type(8))) float v8f;
typedef __attribute__((ext_vector_type(4))) int   i4v;

#define Hdim 512
#define Ldim 2048
#define Bdim 16
#define Ndim 64
#define Tdim 64
#define Jdim 32
#define USTR 68              // padded LDS stride for U[b][t] (bank spread, 16B mult)
#define TABSTRIDE 20608      // 5*4096 + 128 floats per h

#if defined(__AMDGCN__) && __has_builtin(__builtin_amdgcn_global_load_async_to_lds_b128)
#define HAVE_ASYNC 1
#else
#define HAVE_ASYNC 0
#endif

#if HAVE_ASYNC
typedef __attribute__((address_space(1))) i4v* g_v4p;  // global 16B vector ptr
typedef __attribute__((address_space(3))) i4v* l_v4p;  // LDS 16B vector ptr
// 16-byte async copy global -> LDS (tracked by ASYNCcnt, no VGPR staging).
__device__ __forceinline__ void async_cp16(const float* g, float* l) {
  __builtin_amdgcn_global_load_async_to_lds_b128(
      (g_v4p)(unsigned long long)(uintptr_t)g,
      (l_v4p)(unsigned)(uintptr_t)l, 0, 0);
}
__device__ __forceinline__ void wait_async0() {
#if __has_builtin(__builtin_amdgcn_s_wait_asynccnt)
  __builtin_amdgcn_s_wait_asynccnt(0);
#else
  asm volatile("s_wait_asynccnt 0x0" ::: "memory");
#endif
}
#endif

__device__ __forceinline__ v8f wmma4(v2f a, v2f b, v8f c) {
  // D(16x16,f32) = A(16x4,f32) * B(4x16,f32) + C
  return __builtin_amdgcn_wmma_f32_16x16x4_f32(
      false, a, false, b, (short)0, c, false, false);
}

// A fragment from a K-major table tab[k][m] (ldm = M stride).
// ISA 32-bit A 16x4 layout: lanes 0-15: M=lane, VGPR0=K0,VGPR1=K1;
// lanes 16-31: M=lane-16, VGPR0=K2, VGPR1=K3.
__device__ __forceinline__ v2f loadA_g(const float* __restrict__ tab,
                                       int k0, int m0, int lane, int ldm) {
  int m  = m0 + (lane & 15);
  int kx = k0 + ((lane < 16) ? 0 : 2);
  v2f a;
  a.x = tab[kx * ldm + m];
  a.y = tab[(kx + 1) * ldm + m];
  return a;
}

// B fragment from LDS buffer s[k][16] (row-striped like C/D halves).
__device__ __forceinline__ v2f loadB_s(const float* s, int k0, int lane) {
  int n  = lane & 15;
  int kx = k0 + ((lane < 16) ? 0 : 2);
  v2f b;
  b.x = s[kx * 16 + n];
  b.y = s[(kx + 1) * 16 + n];
  return b;
}

// B fragment from b-major U staging Us[b][t] (stride USTR): B[k=t][n=b].
__device__ __forceinline__ v2f loadB_U(const float* Us, int k0, int lane) {
  int n  = lane & 15;
  int kx = k0 + ((lane < 16) ? 0 : 2);
  v2f b;
  b.x = Us[n * USTR + kx];
  b.y = Us[n * USTR + kx + 1];
  return b;
}

// ---------------------------------------------------------------------------
// Phase 0: per-h tables.
//   KtriT[s][t] = (t>=s) ? k[t-s] : 0        (k[d] = Re(sum_n 2*Cc_n w_n^d))
//   ErT[n][t]   =  Re(2*Cc_n w_n^{t+1})
//   EiT[n][t]   = -Im(2*Cc_n w_n^{t+1})
//   PrT[s][n]   =  Re(w_n^{T-1-s}),  PiT[s][n] = Im(w_n^{T-1-s})
//   wT[n]       =  w_n^T (complex)
// ---------------------------------------------------------------------------
__global__ __launch_bounds__(64) void s4_tables(
    const float* __restrict__ log_dt, const float* __restrict__ A_re,
    const float* __restrict__ A_im, const float* __restrict__ C_re,
    const float* __restrict__ C_im, float* __restrict__ tabs) {
  int h = blockIdx.x;
  int n = threadIdx.x;
  float* tb = tabs + (size_t)h * TABSTRIDE;
  __shared__ float G[64 * 64];
  __shared__ float karr[64];

  float dt  = expf(log_dt[h]);
  float are = -expf(A_re[h * Ndim + n]);
  float aim = A_im[h * Ndim + n];
  float dre = are * dt, dim = aim * dt;
  float ew  = expf(dre);
  float wre = ew * cosf(dim), wim = ew * sinf(dim);
  // (w - 1)/A
  float nre = wre - 1.0f, nim = wim;
  float den = are * are + aim * aim;
  float qre = (nre * are + nim * aim) / den;
  float qim = (nim * are - nre * aim) / den;
  float cre = C_re[h * Ndim + n], cim = C_im[h * Ndim + n];
  float c2r = 2.0f * (cre * qre - cim * qim);  // 2*Cc
  float c2i = 2.0f * (cre * qim + cim * qre);

  float* ErT = tb + 4096;
  float* EiT = tb + 8192;
  float cur_r = c2r, cur_i = c2i;  // 2*Cc * w^0
  for (int t = 0; t < 64; ++t) {
    G[t * 64 + n] = cur_r;  // contribution to k[t]
    float r2 = cur_r * wre - cur_i * wim;
    float i2 = cur_r * wim + cur_i * wre;
    cur_r = r2; cur_i = i2;          // 2*Cc * w^{t+1}
    ErT[n * 64 + t] = cur_r;
    EiT[n * 64 + t] = -cur_i;
  }
  float* PrT = tb + 12288;
  float* PiT = tb + 16384;
  float pr = 1.0f, pi = 0.0f;  // w^0
  for (int p = 0; p < 64; ++p) {
    int s = 63 - p;            // power T-1-s = p
    PrT[s * 64 + n] = pr;
    PiT[s * 64 + n] = pi;
    float r2 = pr * wre - pi * wim;
    float i2 = pr * wim + pi * wre;
    pr = r2; pi = i2;
  }
  tb[20480 + n] = pr;  // Re(w^64)
  tb[20544 + n] = pi;  // Im(w^64)
  __syncthreads();
  float kv = 0.0f;
  for (int m = 0; m < 64; ++m) kv += G[n * 64 + m];
  karr[n] = kv;
  __syncthreads();
  float* Ktri = tb;  // [s][t]
  int s = n;
  for (int t = 0; t < 64; ++t)
    Ktri[s * 64 + t] = (t >= s) ? karr[t - s] : 0.0f;
}

// ---------------------------------------------------------------------------
// Transpose W_out[v][u] -> WT[u][v] for coalesced A-fragment loads.
// ---------------------------------------------------------------------------
__global__ void s4_transposeW(const float* __restrict__ W,
                              float* __restrict__ WT) {
  int idx = blockIdx.x * 256 + threadIdx.x;  // 512*512
  int v = idx >> 9, hh = idx & 511;
  WT[hh * 512 + v] = W[idx];
}

// ---------------------------------------------------------------------------
// Phase A: one block per h. Sequential over 32 chunks, batched over B=16.
// Per chunk: Y = Ktri@U + Er@XR + Ei@XI ; S = P@U ; X <- w^T X + S.
// Then y += D*u, exact GELU, store yact. 4 waves: wave w owns M-rows 16w..16w+15.
// ---------------------------------------------------------------------------
__global__ __launch_bounds__(128) void s4_phaseA(
    const float* __restrict__ u, const float* __restrict__ D,
    const float* __restrict__ tabs, float* __restrict__ yact) {
  int h = blockIdx.x;
  int tid = threadIdx.x, lane = tid & 31, wv = tid >> 5;
  const float* tb   = tabs + (size_t)h * TABSTRIDE;
  const float* Ktri = tb;           // [s][t]
  const float* ErT  = tb + 4096;    // [n][t]
  const float* EiT  = tb + 8192;    // [n][t]
  const float* PrT  = tb + 12288;   // [s][n]
  const float* PiT  = tb + 16384;   // [s][n]

  __shared__ float Us[Bdim * USTR];               // U[b][t], padded
  __shared__ float XR[Ndim * 16], XI[Ndim * 16];  // state  [n][b]
  __shared__ float SRs[Ndim * 16], SIs[Ndim * 16];
  __shared__ float wtr[64], wti[64];

  for (int i = tid; i < Ndim * 16; i += 128) { XR[i] = 0.0f; XI[i] = 0.0f; }
  if (tid < 64) { wtr[tid] = tb[20480 + tid]; wti[tid] = tb[20544 + tid]; }
  float Dh = D[h];
  int m0 = wv * 16;

  for (int j = 0; j < Jdim; ++j) {
    __syncthreads();
    // Stage U[b][t] into LDS: 16 rows x 64 floats.
#if HAVE_ASYNC
    for (int r = 0; r < 2; ++r) {
      int o = tid + r * 128;   // 256 x 16B transfers
      int b = o >> 4, tq = o & 15;
      async_cp16(u + ((size_t)(b * Hdim + h)) * Ldim + j * Tdim + tq * 4,
                 &Us[b * USTR + tq * 4]);
    }
    wait_async0();
#else
    for (int r = 0; r < 8; ++r) {
      int o = tid + r * 128;
      int t = o & 63, b = o >> 6;
      Us[b * USTR + t] = u[((size_t)(b * Hdim + h)) * Ldim + j * Tdim + t];
    }
#endif
    __syncthreads();

    v8f acc = {};
    for (int kb = 0; kb < 16; ++kb)
      acc = wmma4(loadA_g(Ktri, kb * 4, m0, lane, 64), loadB_U(Us, kb * 4, lane), acc);
    for (int kb = 0; kb < 16; ++kb)
      acc = wmma4(loadA_g(ErT, kb * 4, m0, lane, 64), loadB_s(XR, kb * 4, lane), acc);
    for (int kb = 0; kb < 16; ++kb)
      acc = wmma4(loadA_g(EiT, kb * 4, m0, lane, 64), loadB_s(XI, kb * 4, lane), acc);

    v8f sr = {}, si = {};
    for (int kb = 0; kb < 16; ++kb) {
      v2f bu = loadB_U(Us, kb * 4, lane);
      sr = wmma4(loadA_g(PrT, kb * 4, m0, lane, 64), bu, sr);
      si = wmma4(loadA_g(PiT, kb * 4, m0, lane, 64), bu, si);
    }

    // Scatter S to LDS; apply skip + exact GELU; write yact.
    int col = lane & 15;
    int rhi = (lane < 16) ? 0 : 8;
    for (int p = 0; p < 8; ++p) {
      int row = m0 + rhi + p;  // t (for Y) == n (for S), both 0..63
      SRs[row * 16 + col] = sr[p];
      SIs[row * 16 + col] = si[p];
      float uval = Us[col * USTR + row];
      float yv = acc[p] + Dh * uval;
      float g = 0.5f * yv * (1.0f + erff(yv * 0.70710678118f));
      yact[((size_t)(col * Hdim + h)) * Ldim + j * Tdim + row] = g;
    }
    __syncthreads();

    // State update X <- w^T * X + S (complex, per (n,b)).
    for (int r = 0; r < 8; ++r) {
      int idx = tid + r * 128;
      int n = idx >> 4;
      float xr = XR[idx], xi = XI[idx];
      float nr = wtr[n] * xr - wti[n] * xi + SRs[idx];
      float ni = wti[n] * xr + wtr[n] * xi + SIs[idx];
      XR[idx] = nr; XI[idx] = ni;
    }
  }
}

// ---------------------------------------------------------------------------
// Phase B: out[b,v,l] = sum_h WT[h][v]*yact[b,h,l] + bias[v].
// Block = 64v x 16l tile: 4 waves share one async-staged yact K-panel
// (512x16 = 32KB LDS); wave wv owns v-rows v0..v0+15. K=512 in 128 wmma.
// 16384 blocks = 16 b * 8 vgroups * 128 ltiles.
// ---------------------------------------------------------------------------
__global__ __launch_bounds__(128) void s4_out(
    const float* __restrict__ WT, const float* __restrict__ yact,
    const float* __restrict__ bias, float* __restrict__ out) {
  int tid = threadIdx.x, lane = tid & 31, wv = tid >> 5;
  int blk = blockIdx.x;
  int b   = blk >> 10;          // 8 vgroups * 128 ltiles per b
  int rem = blk & 1023;
  int vg  = rem >> 7, lt = rem & 127;
  int l0  = lt * 16;
  int v0  = vg * 64 + wv * 16;

  __shared__ float Yl[512 * 16];  // yact[k=0..511][l0..l0+15]
  const float* yb = yact + (size_t)b * Hdim * Ldim + l0;

#if HAVE_ASYNC
  for (int r = 0; r < 16; ++r) {
    int o = tid + r * 128;       // 2048 x 16B transfers
    int k = o >> 2, q = o & 3;
    async_cp16(yb + (size_t)k * Ldim + q * 4, &Yl[k * 16 + q * 4]);
  }
  wait_async0();
#else
  for (int r = 0; r < 64; ++r) {
    int o = tid + r * 128;
    int k = o >> 4, c = o & 15;
    Yl[o] = yb[(size_t)k * Ldim + c];
  }
#endif
  __syncthreads();

  v8f acc = {};
  int mn = lane & 15;
  for (int kb = 0; kb < 128; ++kb) {
    int k0 = kb * 4;
    int kx = k0 + ((lane < 16) ? 0 : 2);
    v2f a, bb;
    a.x  = WT[kx * 512 + v0 + mn];
    a.y  = WT[(kx + 1) * 512 + v0 + mn];
    bb.x = Yl[kx * 16 + mn];
    bb.y = Yl[(kx + 1) * 16 + mn];
    acc = wmma4(a, bb, acc);
  }
  int col = l0 + mn;
  int rhi = (lane < 16) ? 0 : 8;
  for (int p = 0; p < 8; ++p) {
    int v = v0 + rhi + p;
    out[((size_t)(b * Hdim + v)) * Ldim + col] = acc[p] + bias[v];
  }
}

// ---------------------------------------------------------------------------
extern "C" void kernel_launch(void* const* d_in, const int* in_sizes, int n_in,
                              void* d_out, int out_size, void* d_ws, size_t ws_size,
                              hipStream_t stream) {
  const float* u      = (const float*)d_in[0];
  const float* log_dt = (const float*)d_in[1];
  const float* A_re   = (const float*)d_in[2];
  const float* A_im   = (const float*)d_in[3];
  const float* C_re   = (const float*)d_in[4];
  const float* C_im   = (const float*)d_in[5];
  const float* Dm     = (const float*)d_in[6];
  const float* W_out  = (const float*)d_in[7];
  const float* b_out  = (const float*)d_in[8];
  float* out = (float*)d_out;

  float* ws   = (float*)d_ws;
  float* yact = ws;                                   // 16,777,216 floats
  float* WT   = ws + 16777216;                        //    262,144 floats
  float* tabs = ws + 16777216 + 262144;               // 512*20608 floats

  s4_tables<<<Hdim, 64, 0, stream>>>(log_dt, A_re, A_im, C_re, C_im, tabs);
  s4_transposeW<<<1024, 256, 0, stream>>>(W_out, WT);
  s4_phaseA<<<Hdim, 128, 0, stream>>>(u, Dm, tabs, yact);
  s4_out<<<16384, 128, 0, stream>>>(WT, yact, b_out, out);
}